// ListMLE_50294067036268
// MI455X (gfx1250) — compile-verified
//
#include <hip/hip_runtime.h>
#include <math.h>

#define NCOLS 4096
#define TPB   256
#define EPT   (NCOLS / TPB)   // 16 elements per thread

typedef int v4i __attribute__((vector_size(16)));
typedef __attribute__((address_space(1))) v4i* GV4;
typedef __attribute__((address_space(3))) v4i* LV4;

struct MS { float m; float s; };

union ShBuf {
  unsigned long long pairs[NCOLS];  // 32 KB: packed (key<<32)|payload
  float f[2 * NCOLS];               // same 32 KB: raw staging (labels | outputs)
};

__global__ __launch_bounds__(TPB) void listmle_rows(
    const float* __restrict__ outputs,
    const float* __restrict__ labels,
    float* __restrict__ row_partials)
{
  __shared__ ShBuf sh;
  __shared__ MS    agg[TPB];
  __shared__ float red[TPB];

  const int    tid  = threadIdx.x;
  const size_t base = (size_t)blockIdx.x * NCOLS;
  const float* lsrc = labels  + base;
  const float* osrc = outputs + base;

  // ---- Stage labels -> sh.f[0..N), outputs -> sh.f[N..2N) ----
#if defined(__gfx1250__) && __has_builtin(__builtin_amdgcn_global_load_async_to_lds_b128)
  // CDNA5 async copy engine: each lane moves 16B; 4 rounds x 2 arrays covers 32 KB.
  #pragma unroll
  for (int c = 0; c < 4; ++c) {
    const int idx = (c * TPB + tid) * 4;  // float index, 16B aligned chunks
    __builtin_amdgcn_global_load_async_to_lds_b128(
        (GV4)(lsrc + idx), (LV4)&sh.f[idx], 0, 0);
    __builtin_amdgcn_global_load_async_to_lds_b128(
        (GV4)(osrc + idx), (LV4)&sh.f[NCOLS + idx], 0, 0);
  }
  #if __has_builtin(__builtin_amdgcn_s_wait_asynccnt)
  __builtin_amdgcn_s_wait_asynccnt(0);
  #else
  asm volatile("s_wait_asynccnt 0x0" ::: "memory");
  #endif
#else
  #pragma unroll
  for (int c = 0; c < EPT; ++c) {
    const int i = c * TPB + tid;
    sh.f[i]         = lsrc[i];
    sh.f[NCOLS + i] = osrc[i];
  }
#endif
  __syncthreads();

  // ---- Pack (label-key, output-payload) u64 in place over the staging ----
  float labv[EPT], outv[EPT];
  float outsum = 0.0f;
  #pragma unroll
  for (int c = 0; c < EPT; ++c) {
    const int i = c * TPB + tid;
    labv[c] = sh.f[i];
    outv[c] = sh.f[NCOLS + i];
    outsum += outv[c];
  }
  __syncthreads();  // everyone has read raw floats before u64 overwrite
  #pragma unroll
  for (int c = 0; c < EPT; ++c) {
    const int i = c * TPB + tid;
    unsigned kb = __float_as_uint(labv[c]);
    kb = (kb & 0x80000000u) ? ~kb : (kb | 0x80000000u);  // float -> ascending uint order
    sh.pairs[i] = ((unsigned long long)kb << 32) |
                  (unsigned long long)__float_as_uint(outv[c]);
  }
  __syncthreads();

  // ---- Bitonic sort (ascending) of 4096 u64 pairs in LDS ----
  for (unsigned k = 2; k <= (unsigned)NCOLS; k <<= 1) {
    for (unsigned j = k >> 1; j > 0; j >>= 1) {
      #pragma unroll
      for (int c = 0; c < EPT; ++c) {
        const unsigned i   = (unsigned)(c * TPB + tid);
        const unsigned ixj = i ^ j;
        if (ixj > i) {
          const unsigned long long a = sh.pairs[i];
          const unsigned long long b = sh.pairs[ixj];
          const bool up = ((i & k) == 0);
          if (up ? (a > b) : (a < b)) {
            sh.pairs[i]   = b;
            sh.pairs[ixj] = a;
          }
        }
      }
      __syncthreads();
    }
  }

  // ---- Inclusive LSE-scan: local sequential scan over contiguous chunk ----
  float m[EPT], s[EPT];
  {
    float rm = -__builtin_inff(), rs = 0.0f;
    #pragma unroll
    for (int c = 0; c < EPT; ++c) {
      const float v = __uint_as_float(
          (unsigned)(sh.pairs[tid * EPT + c] & 0xFFFFFFFFull));
      const float M = fmaxf(rm, v);
      rs = rs * __expf(rm - M) + __expf(v - M);
      rm = M;
      m[c] = rm;
      s[c] = rs;
    }
    agg[tid].m = rm;
    agg[tid].s = rs;
  }
  __syncthreads();

  // ---- Hillis-Steele inclusive scan of the 256 chunk aggregates ----
  for (int off = 1; off < TPB; off <<= 1) {
    MS x = agg[tid];
    MS y;
    const bool have = (tid >= off);
    if (have) y = agg[tid - off];
    __syncthreads();
    if (have) {
      const float M = fmaxf(y.m, x.m);
      MS r;
      r.s = y.s * __expf(y.m - M) + x.s * __expf(x.m - M);
      r.m = M;
      agg[tid] = r;
    }
    __syncthreads();
  }

  // ---- Combine exclusive prefix with local scan; sum the scores ----
  float exm = -__builtin_inff(), exs = 0.0f;
  if (tid > 0) { exm = agg[tid - 1].m; exs = agg[tid - 1].s; }
  float ssum = 0.0f;
  #pragma unroll
  for (int c = 0; c < EPT; ++c) {
    const float M = fmaxf(exm, m[c]);
    const float S = exs * __expf(exm - M) + s[c] * __expf(m[c] - M);
    ssum += M + __logf(S);
  }

  // ---- Row reduction: sum(scores) - sum(outputs) ----
  red[tid] = ssum - outsum;
  __syncthreads();
  for (int off = TPB / 2; off > 0; off >>= 1) {
    if (tid < off) red[tid] += red[tid + off];
    __syncthreads();
  }
  if (tid == 0) row_partials[blockIdx.x] = red[0];
}

__global__ __launch_bounds__(TPB) void listmle_final(
    const float* __restrict__ row_partials, int nrows,
    float* __restrict__ out, double inv_count)
{
  __shared__ double red[TPB];
  double acc = 0.0;
  for (int i = threadIdx.x; i < nrows; i += TPB) acc += (double)row_partials[i];
  red[threadIdx.x] = acc;
  __syncthreads();
  for (int off = TPB / 2; off > 0; off >>= 1) {
    if (threadIdx.x < off) red[threadIdx.x] += red[threadIdx.x + off];
    __syncthreads();
  }
  if (threadIdx.x == 0) out[0] = (float)(red[0] * inv_count);
}

extern "C" void kernel_launch(void* const* d_in, const int* in_sizes, int n_in,
                              void* d_out, int out_size, void* d_ws, size_t ws_size,
                              hipStream_t stream) {
  const float* outputs = (const float*)d_in[0];
  const float* labels  = (const float*)d_in[1];
  const int total = in_sizes[0];          // B * N
  const int B     = total / NCOLS;        // N = 4096 per the reference

  float* row_partials = (float*)d_ws;     // B floats of scratch

  listmle_rows<<<B, TPB, 0, stream>>>(outputs, labels, row_partials);
  listmle_final<<<1, TPB, 0, stream>>>(row_partials, B, (float*)d_out,
                                       1.0 / (double)total);
}